// SVPFModel_2602750181460
// MI455X (gfx1250) — compile-verified
//
#include <hip/hip_runtime.h>
#include <math.h>

#define NB 8            // batches
#define NP 2048         // particles
#define ND 8            // dims
#define HBINS 4096
#define LRATE 0.1f
#define ALPHA 0.9f
#define EPSI 1e-8f
#define LOG2E 1.4426950408889634f

typedef float v2f __attribute__((ext_vector_type(2)));
typedef float v8f __attribute__((ext_vector_type(8)));

// ---------------- histogram / median utilities ----------------

__global__ __launch_bounds__(256) void zero_u32(unsigned int* __restrict__ p, int n) {
  int i = blockIdx.x * 256 + threadIdx.x;
  if (i < n) p[i] = 0u;
}

__global__ __launch_bounds__(256) void norms_kernel(const float* __restrict__ x,
                                                    float* __restrict__ nrm) {
  int i = blockIdx.x * 256 + threadIdx.x;
  if (i < NB * NP) {
    const float* p = x + i * ND;
    float s = 0.f;
#pragma unroll
    for (int d = 0; d < ND; ++d) s += p[d] * p[d];
    nrm[i] = s;
  }
}

// pairwise-distance histogram; pass0: [0,32) coarse, pass1: per-batch window from sel
__global__ __launch_bounds__(256) void hist_kernel(const float* __restrict__ x,
                                                   const float* __restrict__ nrm,
                                                   unsigned int* __restrict__ hist,
                                                   const float* __restrict__ sel,
                                                   int pass) {
  __shared__ unsigned int lh[HBINS];
  for (int t = threadIdx.x; t < HBINS; t += 256) lh[t] = 0u;
  __syncthreads();

  int row = blockIdx.x * 256 + threadIdx.x;   // 8 blocks per batch -> one batch per block
  int b = row / NP;
  float lo, invw;
  if (pass == 0) { lo = 0.f; invw = (float)HBINS / 32.0f; }
  else           { lo = sel[b * 4 + 0]; invw = 1.0f / sel[b * 4 + 1]; }

  float xi[ND];
  const float* pr = x + row * ND;
#pragma unroll
  for (int d = 0; d < ND; ++d) xi[d] = pr[d];
  float ni = nrm[row];
  const float* xb  = x   + b * NP * ND;
  const float* nbm = nrm + b * NP;

  for (int j = 0; j < NP; ++j) {
    const float* pj = xb + j * ND;
    float dot = 0.f;
#pragma unroll
    for (int d = 0; d < ND; ++d) dot += xi[d] * pj[d];
    float sq = ni + nbm[j] - 2.0f * dot;
    float dist = sqrtf(fmaxf(sq, 0.f));
    int bin = (int)floorf((dist - lo) * invw);
    if (pass == 0) bin = bin > HBINS - 1 ? HBINS - 1 : bin;   // clamp far tail
    if (bin >= 0 && bin < HBINS) atomicAdd(&lh[bin], 1u);
  }
  __syncthreads();
  unsigned int* gh = hist + b * HBINS;
  for (int t = threadIdx.x; t < HBINS; t += 256) {
    unsigned int v = lh[t];
    if (v) atomicAdd(&gh[t], v);
  }
}

__global__ void select_kernel(const unsigned int* __restrict__ hist,
                              float* __restrict__ sel,
                              float* __restrict__ hout, int pass) {
  int b = threadIdx.x;
  if (b >= NB) return;
  const unsigned int* gh = hist + b * HBINS;
  if (pass == 0) {
    const unsigned int rank = (unsigned int)NP * (unsigned int)NP / 2u;
    unsigned int cum = 0, before = 0;
    int bin = HBINS - 1;
    for (int t = 0; t < HBINS; ++t) {
      unsigned int c = gh[t];
      if (cum + c >= rank) { bin = t; before = cum; break; }
      cum += c;
    }
    const float w0 = 32.0f / (float)HBINS;
    sel[b * 4 + 0] = (float)bin * w0;          // window lo
    sel[b * 4 + 1] = w0 / (float)HBINS;        // fine bin width
    sel[b * 4 + 2] = (float)(rank - before);   // remaining rank (exact in fp32, < 2^24)
  } else {
    unsigned int r1 = (unsigned int)sel[b * 4 + 2];
    unsigned int cum = 0;
    int bin = HBINS - 1;
    for (int t = 0; t < HBINS; ++t) {
      unsigned int c = gh[t];
      if (cum + c >= r1) { bin = t; break; }
      cum += c;
    }
    float med = sel[b * 4 + 0] + ((float)bin + 0.5f) * sel[b * 4 + 1];
    hout[b] = med * med / logf((float)NP);
  }
}

// ---------------- main SVGD iteration: fp32 WMMA ----------------
// One wave32 per (batch, 16-row strip). Gram tile via V_WMMA_F32_16X16X4_F32,
// K = exp2(S*2*log2e/h - sn_i - sn_j), then P += K @ M' via 4 more WMMAs per
// j-tile, with M' = [obs - (1+2/h)x_j | 1 | 0...], so P[:,8] is the K-rowsum.
__global__ __launch_bounds__(32) void svgd_wmma_kernel(const float* __restrict__ xin,
                                                       float* __restrict__ xout,
                                                       float* __restrict__ sst,
                                                       const float* __restrict__ nrm,
                                                       const float* __restrict__ hbuf,
                                                       const float* __restrict__ obs,
                                                       int first) {
  __shared__ float ldsK[256];
  __shared__ float ldsM[256];
  __shared__ float ldsP[256];

  const int strip = blockIdx.x;         // 0 .. NB*(NP/16)-1
  const int b  = strip >> 7;            // NP/16 == 128
  const int i0 = (strip & 127) << 4;
  const int lane = threadIdx.x;
  const int g = lane >> 4;              // half-wave
  const int c = lane & 15;

  const float h     = hbuf[b];
  const float inv_h = 1.0f / h;
  const float t2h   = 2.0f * inv_h;
  const float m1    = 1.0f + t2h;
  const float l2h   = LOG2E * inv_h;    // log2(e)/h : exp(-sq/h) == exp2(-sq*l2h)
  const float t2l   = 2.0f * l2h;

  const float* xb  = xin + b * NP * ND;
  const float* nbm = nrm + b * NP;

  // obs at this lane's four dims {2g, 2g+1, 2g+4, 2g+5} (loop-invariant)
  const float ob0 = obs[b * ND + 2 * g + 0];
  const float ob1 = obs[b * ND + 2 * g + 1];
  const float ob2 = obs[b * ND + 2 * g + 4];
  const float ob3 = obs[b * ND + 2 * g + 5];

  // constant columns of M': col 8 = ones (K row-sum), cols 9..15 = 0. Written once.
#pragma unroll
  for (int e = 0; e < 4; ++e) {
    int col = 8 + 4 * g + e;
    ldsM[c * 16 + col] = (col == 8) ? 1.0f : 0.0f;
  }

  // A operand of Gram (x_i strip), invariant over j loop.
  // A(m,k): lane = m + 16*(k>>1), vgpr = k&1  ->  lane: m=c, k = 2g+v (+4 for a1)
  v2f a0, a1;
  {
    const float* pri = xb + (i0 + c) * ND;
    a0.x = pri[2 * g + 0]; a0.y = pri[2 * g + 1];
    a1.x = pri[2 * g + 4]; a1.y = pri[2 * g + 5];
  }
  // prescaled row norms: sni[r] = |x_i|^2 * log2e / h
  float sni[8];
#pragma unroll
  for (int r = 0; r < 8; ++r) sni[r] = nbm[i0 + 8 * g + r] * l2h;

  v8f acc = {0.f, 0.f, 0.f, 0.f, 0.f, 0.f, 0.f, 0.f};   // P tile (C layout)

  for (int j0 = 0; j0 < NP; j0 += 16) {
    // B operand of Gram = X_j^T: B(k,n)=x[j0+n, k] -> same gather pattern as A
    const float* prj = xb + (j0 + c) * ND;
    v2f b0, b1;
    b0.x = prj[2 * g + 0]; b0.y = prj[2 * g + 1];
    b1.x = prj[2 * g + 4]; b1.y = prj[2 * g + 5];
    if (j0 + 16 < NP) __builtin_prefetch(xb + (j0 + 16 + c) * ND, 0, 3);

    v8f sacc = {0.f, 0.f, 0.f, 0.f, 0.f, 0.f, 0.f, 0.f};
    sacc = __builtin_amdgcn_wmma_f32_16x16x4_f32(false, a0, false, b0, (short)0, sacc, false, false);
    sacc = __builtin_amdgcn_wmma_f32_16x16x4_f32(false, a1, false, b1, (short)0, sacc, false, false);

    // C layout: element (m = 8g+r, n = c) in sacc[r]. arg <= 0 always.
    float snj = nbm[j0 + c] * l2h;
#pragma unroll
    for (int r = 0; r < 8; ++r) {
      float kv = __builtin_amdgcn_exp2f(fmaf(sacc[r], t2l, -(sni[r] + snj)));
      ldsK[(8 * g + r) * 16 + c] = kv;   // exp(-|xi-xj|^2 / h)
    }
    // M' tile (rows j0..j0+15, dims 0..7) built straight from the B registers:
    // this lane owns row (j0+c), dims {2g,2g+1,2g+4,2g+5} -> full 16x8 coverage.
    ldsM[c * 16 + 2 * g + 0] = ob0 - m1 * b0.x;
    ldsM[c * 16 + 2 * g + 1] = ob1 - m1 * b0.y;
    ldsM[c * 16 + 2 * g + 4] = ob2 - m1 * b1.x;
    ldsM[c * 16 + 2 * g + 5] = ob3 - m1 * b1.y;
    __syncthreads();

    // P += K(16x16) @ M'(16x16), K-dim chunks of 4
#pragma unroll
    for (int kc = 0; kc < 4; ++kc) {
      v2f a2, b2;
      a2.x = ldsK[c * 16 + kc * 4 + 2 * g + 0];              // A2(m=c, k)
      a2.y = ldsK[c * 16 + kc * 4 + 2 * g + 1];
      b2.x = ldsM[(kc * 4 + 2 * g + 0) * 16 + c];            // B2(k, n=c)
      b2.y = ldsM[(kc * 4 + 2 * g + 1) * 16 + c];
      acc = __builtin_amdgcn_wmma_f32_16x16x4_f32(false, a2, false, b2, (short)0, acc, false, false);
    }
    __syncthreads();
  }

  // spill P tile and apply RMSprop update
#pragma unroll
  for (int r = 0; r < 8; ++r) ldsP[(8 * g + r) * 16 + c] = acc[r];
  __syncthreads();

  const float invN = 1.0f / (float)NP;
#pragma unroll
  for (int e = 0; e < 4; ++e) {
    int idx = lane * 4 + e;            // 128 elements: 16 rows x 8 dims
    int m = idx >> 3;
    int d = idx & 7;
    int gi = (b * NP + i0 + m) * ND + d;
    float xold = xin[gi];
    float P    = ldsP[m * 16 + d];
    float rs   = ldsP[m * 16 + 8];     // K row-sum
    float phi  = invN * (P + t2h * xold * rs);
    float sold = first ? 0.0f : sst[gi];
    float snew = ALPHA * sold + (1.0f - ALPHA) * phi * phi;
    sst[gi]  = snew;
    xout[gi] = xold + LRATE * phi / (sqrtf(snew) + EPSI);
  }
}

// ---------------- host-side orchestration ----------------

extern "C" void kernel_launch(void* const* d_in, const int* in_sizes, int n_in,
                              void* d_out, int out_size, void* d_ws, size_t ws_size,
                              hipStream_t stream) {
  const float* x_in = (const float*)d_in[0];   // [8,2048,8]
  const float* obs  = (const float*)d_in[1];   // [8,8]
  float* out = (float*)d_out;                  // [8,2048,8]
  float* ws  = (float*)d_ws;

  float* x0   = ws;                 // 131072 floats
  float* x1   = ws + 131072;        // 131072
  float* sbuf = ws + 262144;        // 131072 (RMSprop state)
  float* nrm  = ws + 393216;        // 16384
  float* hbuf = ws + 409600;        // 8 (bandwidth h per batch)
  float* sel  = ws + 409608;        // 32 (selection state)
  unsigned int* hist = (unsigned int*)(ws + 409640);   // 32768 u32

  // --- bandwidth h from initial x via two-level histogram median ---
  zero_u32<<<NB * HBINS / 256, 256, 0, stream>>>(hist, NB * HBINS);
  norms_kernel<<<NB * NP / 256, 256, 0, stream>>>(x_in, nrm);
  hist_kernel<<<NB * NP / 256, 256, 0, stream>>>(x_in, nrm, hist, sel, 0);
  select_kernel<<<1, 32, 0, stream>>>(hist, sel, hbuf, 0);
  zero_u32<<<NB * HBINS / 256, 256, 0, stream>>>(hist, NB * HBINS);
  hist_kernel<<<NB * NP / 256, 256, 0, stream>>>(x_in, nrm, hist, sel, 1);
  select_kernel<<<1, 32, 0, stream>>>(hist, sel, hbuf, 1);

  // --- 5 SVGD iterations, double-buffered ---
  const float* cur = x_in;
  for (int it = 0; it < 5; ++it) {
    float* nxt = (it == 4) ? out : ((it & 1) ? x1 : x0);
    norms_kernel<<<NB * NP / 256, 256, 0, stream>>>(cur, nrm);
    svgd_wmma_kernel<<<NB * (NP / 16), 32, 0, stream>>>(cur, nxt, sbuf, nrm, hbuf, obs,
                                                        it == 0 ? 1 : 0);
    cur = nxt;
  }
  (void)in_sizes; (void)n_in; (void)out_size; (void)ws_size;
}